// PRCutBatchLoss_15410342658858
// MI455X (gfx1250) — compile-verified
//
#include <hip/hip_runtime.h>

typedef __attribute__((ext_vector_type(2))) float v2f;
typedef __attribute__((ext_vector_type(4))) float v4f;
typedef __attribute__((ext_vector_type(8))) float v8f;

#define A_DIM 8192
#define B_DIM 8192
#define K_DIM 10
#define PADK  16
#define CHUNK 512                    // columns of W per wave
#define NCHUNK (B_DIM / CHUNK)       // 16
#define NSTRIP (A_DIM / 16)          // 512 row-strips of height 16
#define WAVES_PER_BLOCK 8            // 256 threads = 8 wave32

// ---------------------------------------------------------------------------
// Zero the scalar output (harness poisons d_out with 0xAA before timing).
// ---------------------------------------------------------------------------
__global__ void prcut_zero(float* out) { out[0] = 0.0f; }

// ---------------------------------------------------------------------------
// Build padded + row-pair-interleaved P_r in workspace:
//   logical Ppad[b][0..9] = P_r[b][k], Ppad[b][10] = 1.0 (ones column folds
//   row_sum into the WMMA chain), Ppad[b][11..15] = 0.
// Stored as Qpad[b>>1][n][b&1] so the row pair {Ppad[b][n], Ppad[b+1][n]}
// needed by one WMMA's two B VGPRs is 8 contiguous bytes -> one b64 load.
// 8192*16*4B = 512 KB -> lives in the 192 MB L2, reused by every wave.
// ---------------------------------------------------------------------------
__global__ void prcut_pad(const float* __restrict__ P_r, float* __restrict__ Qpad) {
    int idx = blockIdx.x * blockDim.x + threadIdx.x;
    if (idx >= B_DIM * PADK) return;
    int row = idx >> 4;                    // b
    int col = idx & 15;                    // n
    float v = 0.0f;
    if (col < K_DIM)       v = P_r[row * K_DIM + col];
    else if (col == K_DIM) v = 1.0f;
    Qpad[((row >> 1) * PADK + col) * 2 + (row & 1)] = v;
}

// ---------------------------------------------------------------------------
// Main kernel: one wave per (16-row strip, 512-col chunk) of W.
// D = W_tile @ Ppad_tile accumulated with V_WMMA_F32_16X16X4_F32 (full f32).
// Column->K-slot assignment exploits that the b-reduction is order-free:
// lanes 0-15 b128-load W[m, s..s+3], lanes 16-31 load W[m, s+4..s+7];
// WMMA #1 takes (x,y) of each half (cols {s,s+1,s+4,s+5}), WMMA #2 takes
// (z,w) ({s+2,s+3,s+6,s+7}), with B row pairs (single b64 from Qpad) matched
// per half. Afterwards:
//   contrib[m,n] = wpr[m,n]*(1-2*P_l[m,n]) + rowsum[m]*P_l[m,n],
// rowsum[m] = wpr[m,10] (ones column), divide by clusters_p[n], wave-reduce,
// one atomicAdd per wave.
// ---------------------------------------------------------------------------
__global__ __launch_bounds__(256) void prcut_main(
        const float* __restrict__ W,
        const float* __restrict__ P_l,
        const float* __restrict__ Qpad,
        const float* __restrict__ clusters_p,
        float* __restrict__ out) {
    const int lane  = threadIdx.x & 31;
    const int wave  = threadIdx.x >> 5;
    const int gw    = blockIdx.x * WAVES_PER_BLOCK + wave;
    const int strip = gw % NSTRIP;
    const int chunk = gw / NSTRIP;
    const int a0    = strip * 16;          // first row of this wave's strip
    const int b0    = chunk * CHUNK;       // first column of this wave's chunk

    const int n  = lane & 15;              // N index of B/C/D; row index for A
    const int hi = lane >> 4;              // 0 | 1
    const int co = hi * 4;                 // column sub-offset of this half

    const float* wptr = W + (size_t)(a0 + n) * B_DIM + b0 + co;
    // Qpad row-pair base for this half: rows (b0+co+s, b0+co+s+1) at elem n
    // live at Qpad[((b0+co+s)>>1)*PADK*2 + n*2] (b0, s, co all even).
    const float* bptr = Qpad + (size_t)((b0 + co) >> 1) * (PADK * 2) + n * 2;

    v8f c = {};                            // 16x16 f32 accumulator (8 VGPRs)
    for (int s = 0; s < CHUNK; s += 8) {
        __builtin_prefetch(wptr + s + 128, 0, 1);      // global_prefetch_b8
        v4f a  = *(const v4f*)(wptr + s);              // W[m, b0+s+co .. +co+3]
        v2f b1 = *(const v2f*)(bptr + (s >> 1) * (PADK * 2));              // rows s, s+1
        v2f b2 = *(const v2f*)(bptr + (s >> 1) * (PADK * 2) + PADK * 2);   // rows s+2, s+3
        v2f a1; a1.x = a.x; a1.y = a.y;
        v2f a2; a2.x = a.z; a2.y = a.w;
        // 8 args: (neg_a, A, neg_b, B, c_mod, C, reuse_a, reuse_b)
        c = __builtin_amdgcn_wmma_f32_16x16x4_f32(
                false, a1, false, b1, (short)0, c, false, false);
        c = __builtin_amdgcn_wmma_f32_16x16x4_f32(
                false, a2, false, b2, (short)0, c, false, false);
    }

    // ---- combine: D VGPR v holds rows M=v (lanes 0-15) / M=v+8 (lanes 16-31)
    const int   mbase = a0 + hi * 8;
    const int   src   = (lane & 16) | 10;        // lane holding col 10 (rowsum)
    const int   ncl   = (n < K_DIM) ? n : (K_DIM - 1);   // clamped, in-bounds
    const float mask  = (n < K_DIM) ? 1.0f : 0.0f;
    const float* plp  = P_l + (size_t)mbase * K_DIM + ncl;

    float acc = 0.0f;
#pragma unroll
    for (int v = 0; v < 8; ++v) {
        float cv = c[v];                         // wpr[m, n]
        float rs = __shfl(cv, src, 32);          // rowsum[m] (all lanes exec)
        float pl = plp[v * K_DIM];               // unconditional, in-bounds
        acc += cv * (1.0f - 2.0f * pl) + rs * pl;
    }
    acc = acc * mask / clusters_p[ncl];          // zero out pad columns

    // wave32 reduction, then one atomic per wave
#pragma unroll
    for (int off = 16; off > 0; off >>= 1)
        acc += __shfl_xor(acc, off, 32);
    if (lane == 0) atomicAdd(out, acc);
}

// ---------------------------------------------------------------------------
extern "C" void kernel_launch(void* const* d_in, const int* in_sizes, int n_in,
                              void* d_out, int out_size, void* d_ws, size_t ws_size,
                              hipStream_t stream) {
    const float* W   = (const float*)d_in[0];
    const float* P_l = (const float*)d_in[1];
    const float* P_r = (const float*)d_in[2];
    const float* cp  = (const float*)d_in[3];
    float* out  = (float*)d_out;
    float* Qpad = (float*)d_ws;            // 8192*16 floats = 512 KB scratch

    prcut_zero<<<1, 1, 0, stream>>>(out);
    prcut_pad<<<(B_DIM * PADK + 255) / 256, 256, 0, stream>>>(P_r, Qpad);
    prcut_main<<<(NSTRIP * NCHUNK) / WAVES_PER_BLOCK, 256, 0, stream>>>(
        W, P_l, Qpad, cp, out);
}